// GNN_74895639707841
// MI455X (gfx1250) — compile-verified
//
#include <hip/hip_runtime.h>
#include <hip/hip_bf16.h>

// GNN (GIN-style) pipeline for MI455X / gfx1250.
// D=32 GEMMs via V_WMMA_F32_16X16X4_F32 (exact fp32 matrix path), W staged in LDS.
// Edge aggregation = L2 float atomics; segment-max = order-preserving uint atomics.

#define DF 32            // feature dim
#define GOUT 64          // number of graphs
#define BN_EPS 1e-5f

typedef __attribute__((ext_vector_type(2))) float v2f;
typedef __attribute__((ext_vector_type(8))) float v8f;

// ---- order-preserving float <-> uint mapping (for atomic max over floats) ----
static __device__ __forceinline__ unsigned fmap(float f) {
  unsigned u = __float_as_uint(f);
  return (u & 0x80000000u) ? ~u : (u | 0x80000000u);
}
static __device__ __forceinline__ float funmap(unsigned u) {
  return __uint_as_float((u & 0x80000000u) ? (u & 0x7fffffffu) : ~u);
}
#define MAP_NEG_INF 0x007FFFFFu   // fmap(-inf)

// ---------------------------------------------------------------- fills
__global__ void fill_f32_kernel(float* __restrict__ p, long long n, float v) {
  long long i = (long long)blockIdx.x * blockDim.x + threadIdx.x;
  if (i < n) p[i] = v;
}
__global__ void fill_u32_kernel(unsigned* __restrict__ p, long long n, unsigned v) {
  long long i = (long long)blockIdx.x * blockDim.x + threadIdx.x;
  if (i < n) p[i] = v;
}

// ------------------------------------------------- edge scatter-add (GIN agg)
// one lane per (edge, feature): coalesced 128B gather per edge row,
// float atomicAdd (global_atomic_add_f32) into agg[dst] (hits L2; x fits in 192MB L2).
__global__ void edge_agg_kernel(const float* __restrict__ x,
                                const int* __restrict__ ei,
                                float* __restrict__ agg, long long E) {
  long long idx = (long long)blockIdx.x * blockDim.x + threadIdx.x;
  long long e = idx >> 5;
  int d = (int)(idx & 31);
  if (e >= E) return;
  long long src = ei[e];
  long long dst = ei[E + e];
  atomicAdd(&agg[dst * DF + d], x[src * DF + d]);
}

// ------------------------------------------------- WMMA 16x32 tile GEMM core
// Computes rows [row0, row0+16) of (in (+in2)) @ W, W (DFxDF row-major) in LDS.
// c0 accumulates output cols 0..15, c1 cols 16..31.
// A layout (32-bit 16x4): lanes 0-15 hold K={k,k+1}, lanes 16-31 hold K={k+2,k+3}.
// B layout (32-bit 4x16): lane n<16 holds col n, K rows split same way.
template <bool HAS_IN2>
static __device__ __forceinline__ void gemm16x32(const float* __restrict__ in,
                                                 const float* __restrict__ in2,
                                                 const float* __restrict__ Wl,  // LDS
                                                 int row0, int lane,
                                                 v8f& c0, v8f& c1) {
  const int r = lane & 15;
  const int khalf = (lane >> 4) << 1;   // 0 or 2
  v8f zero = {0.f, 0.f, 0.f, 0.f, 0.f, 0.f, 0.f, 0.f};
  c0 = zero;
  c1 = zero;
#pragma unroll
  for (int k4 = 0; k4 < DF; k4 += 4) {
    const int ka = k4 + khalf;
    v2f a = *(const v2f*)(in + (long long)(row0 + r) * DF + ka);   // 8B aligned (ka even)
    if constexpr (HAS_IN2) {
      v2f a2 = *(const v2f*)(in2 + (long long)(row0 + r) * DF + ka);
      a += a2;                                                     // v_pk_add_f32
    }
    v2f b0, b1;
    b0.x = Wl[(ka + 0) * DF + r];
    b0.y = Wl[(ka + 1) * DF + r];
    b1.x = Wl[(ka + 0) * DF + 16 + r];
    b1.y = Wl[(ka + 1) * DF + 16 + r];
    c0 = __builtin_amdgcn_wmma_f32_16x16x4_f32(false, a, false, b0, (short)0, c0, false, false);
    c1 = __builtin_amdgcn_wmma_f32_16x16x4_f32(false, a, false, b1, (short)0, c1, false, false);
  }
}

static __device__ __forceinline__ float elu1(float v) {
  return v > 0.f ? v : expm1f(v);
}

// ---------------------------------------------- conv MLP: out = ELU(BN((in+in2)W+b))
__global__ void conv_mlp_kernel(const float* __restrict__ in, const float* __restrict__ in2,
                                const float* __restrict__ W, const float* __restrict__ b,
                                const float* __restrict__ gamma, const float* __restrict__ beta,
                                const float* __restrict__ mean, const float* __restrict__ var,
                                float* __restrict__ out, int n) {
  __shared__ float Wl[DF * DF];
  ((float4*)Wl)[threadIdx.x] = ((const float4*)W)[threadIdx.x];  // 256 * 16B = 4KB
  __syncthreads();

  const int wave = blockIdx.x * (blockDim.x >> 5) + (threadIdx.x >> 5);
  const int row0 = wave * 16;
  if (row0 >= n) return;                  // wave-uniform exit: EXEC all-1s for WMMA
  const int lane = threadIdx.x & 31;

  v8f c0, c1;
  gemm16x32<true>(in, in2, Wl, row0, lane, c0, c1);

  const int colA = lane & 15;
  const int colB = colA + 16;
  const float sA = gamma[colA] * rsqrtf(var[colA] + BN_EPS);
  const float tA = (b[colA] - mean[colA]) * sA + beta[colA];
  const float sB = gamma[colB] * rsqrtf(var[colB] + BN_EPS);
  const float tB = (b[colB] - mean[colB]) * sB + beta[colB];
  const int rowoff = (lane >> 4) * 8;     // C layout: lanes 16-31 hold M = r+8
#pragma unroll
  for (int rr = 0; rr < 8; ++rr) {
    const int row = row0 + rr + rowoff;
    out[(long long)row * DF + colA] = elu1(c0[rr] * sA + tA);
    out[(long long)row * DF + colB] = elu1(c1[rr] * sB + tB);
  }
}

// --------------- lin MLP: h=ELU(BN(in W+b)); Z += w*h; segmax(maybe w*h) per graph
__global__ void lin_mlp_kernel(const float* __restrict__ in,
                               const float* __restrict__ W, const float* __restrict__ b,
                               const float* __restrict__ gamma, const float* __restrict__ beta,
                               const float* __restrict__ mean, const float* __restrict__ var,
                               const float* __restrict__ wptr, const int* __restrict__ batch,
                               float* __restrict__ Z, unsigned* __restrict__ mapped,
                               int n, int use_w_for_max) {
  __shared__ float Wl[DF * DF];
  ((float4*)Wl)[threadIdx.x] = ((const float4*)W)[threadIdx.x];
  __syncthreads();

  const int wave = blockIdx.x * (blockDim.x >> 5) + (threadIdx.x >> 5);
  const int row0 = wave * 16;
  if (row0 >= n) return;
  const int lane = threadIdx.x & 31;

  v8f c0, c1;
  gemm16x32<false>(in, nullptr, Wl, row0, lane, c0, c1);

  const float w = *wptr;
  const int colA = lane & 15;
  const int colB = colA + 16;
  const float sA = gamma[colA] * rsqrtf(var[colA] + BN_EPS);
  const float tA = (b[colA] - mean[colA]) * sA + beta[colA];
  const float sB = gamma[colB] * rsqrtf(var[colB] + BN_EPS);
  const float tB = (b[colB] - mean[colB]) * sB + beta[colB];
  const int rowoff = (lane >> 4) * 8;
#pragma unroll
  for (int rr = 0; rr < 8; ++rr) {
    const int row = row0 + rr + rowoff;
    const float hA = elu1(c0[rr] * sA + tA);
    const float hB = elu1(c1[rr] * sB + tB);
    const float zA = w * hA;
    const float zB = w * hB;
    Z[(long long)row * DF + colA] += zA;   // each element owned by exactly one lane
    Z[(long long)row * DF + colB] += zB;
    const unsigned g = (unsigned)batch[row];
    const float mA = use_w_for_max ? zA : hA;  // layer 0 maxes unscaled z
    const float mB = use_w_for_max ? zB : hB;
    atomicMax(&mapped[g * DF + colA], fmap(mA));
    atomicMax(&mapped[g * DF + colB], fmap(mB));
  }
}

// -------------------------------------------- out += unmap(max); reset buckets
__global__ void segmax_finalize_kernel(float* __restrict__ out, unsigned* __restrict__ mapped, int n) {
  int i = blockIdx.x * blockDim.x + threadIdx.x;
  if (i < n) {
    out[i] += funmap(mapped[i]);
    mapped[i] = MAP_NEG_INF;
  }
}

extern "C" void kernel_launch(void* const* d_in, const int* in_sizes, int n_in,
                              void* d_out, int out_size, void* d_ws, size_t ws_size,
                              hipStream_t stream) {
  const int N = in_sizes[0] / DF;               // 100000 (divisible by 16)
  const long long E = in_sizes[1] / 2;          // 1600000

  const float* x_in  = (const float*)d_in[0];
  const int*   ei    = (const int*)d_in[1];
  const int*   batch = (const int*)d_in[2];
  const float* lw    = (const float*)d_in[3];
  const float* lin_W = (const float*)d_in[4];
  const float* lin_b = (const float*)d_in[5];
  const float* lin_g = (const float*)d_in[6];
  const float* lin_e = (const float*)d_in[7];
  const float* lin_m = (const float*)d_in[8];
  const float* lin_v = (const float*)d_in[9];
  const float* cv_W  = (const float*)d_in[10];
  const float* cv_b  = (const float*)d_in[11];
  const float* cv_g  = (const float*)d_in[12];
  const float* cv_e  = (const float*)d_in[13];
  const float* cv_m  = (const float*)d_in[14];
  const float* cv_v  = (const float*)d_in[15];

  const long long ND = (long long)N * DF;
  float* out_p = (float*)d_out;                 // (G, D)
  float* Z_p   = out_p + (long long)GOUT * DF;  // (N, D)
  float* x_p   = Z_p + ND;                      // (N, D) final x

  float* agg       = (float*)d_ws;              // N*D
  float* xcur      = agg + ND;                  // N*D
  unsigned* mapped = (unsigned*)(xcur + ND);    // G*D

  const int GD = GOUT * DF;
  const int fillGridND = (int)((ND + 255) / 256);
  const int fillGridGD = (GD + 255) / 256;
  const int waves = (N + 15) / 16;
  const int mlpGrid = (waves + 7) / 8;          // 256 threads = 8 waves/block
  const long long eThreads = E * DF;
  const int edgeGrid = (int)((eThreads + 255) / 256);

  // init accumulators (d_out/d_ws are poisoned; re-init every call)
  fill_f32_kernel<<<fillGridND, 256, 0, stream>>>(Z_p, ND, 0.f);
  fill_f32_kernel<<<fillGridGD, 256, 0, stream>>>(out_p, GD, 0.f);
  fill_u32_kernel<<<fillGridGD, 256, 0, stream>>>(mapped, GD, MAP_NEG_INF);
  fill_f32_kernel<<<fillGridND, 256, 0, stream>>>(agg, ND, 0.f);

  // ---- layer 0: z = mlp(x; lin0); Z = w0*z; out = segmax(z)
  lin_mlp_kernel<<<mlpGrid, 256, 0, stream>>>(x_in, lin_W, lin_b, lin_g, lin_e, lin_m, lin_v,
                                              lw + 0, batch, Z_p, mapped, N, /*use_w_for_max=*/0);
  segmax_finalize_kernel<<<fillGridGD, 256, 0, stream>>>(out_p, mapped, GD);

  // ---- layer 1: x1 = mlp(x + agg(x); conv0); z = w1*mlp(x1; lin1)
  edge_agg_kernel<<<edgeGrid, 256, 0, stream>>>(x_in, ei, agg, E);
  conv_mlp_kernel<<<mlpGrid, 256, 0, stream>>>(x_in, agg, cv_W, cv_b, cv_g, cv_e, cv_m, cv_v,
                                               xcur, N);
  fill_f32_kernel<<<fillGridND, 256, 0, stream>>>(agg, ND, 0.f);
  lin_mlp_kernel<<<mlpGrid, 256, 0, stream>>>(xcur,
                                              lin_W + DF * DF, lin_b + DF, lin_g + DF, lin_e + DF,
                                              lin_m + DF, lin_v + DF,
                                              lw + 1, batch, Z_p, mapped, N, 1);
  segmax_finalize_kernel<<<fillGridGD, 256, 0, stream>>>(out_p, mapped, GD);

  // ---- layer 2: x2 = mlp(x1 + agg(x1); conv1) -> d_out x slot; z = w2*mlp(x2; lin2)
  edge_agg_kernel<<<edgeGrid, 256, 0, stream>>>(xcur, ei, agg, E);
  conv_mlp_kernel<<<mlpGrid, 256, 0, stream>>>(xcur, agg,
                                               cv_W + DF * DF, cv_b + DF, cv_g + DF, cv_e + DF,
                                               cv_m + DF, cv_v + DF,
                                               x_p, N);
  lin_mlp_kernel<<<mlpGrid, 256, 0, stream>>>(x_p,
                                              lin_W + 2 * DF * DF, lin_b + 2 * DF, lin_g + 2 * DF,
                                              lin_e + 2 * DF, lin_m + 2 * DF, lin_v + 2 * DF,
                                              lw + 2, batch, Z_p, mapped, N, 1);
  segmax_finalize_kernel<<<fillGridGD, 256, 0, stream>>>(out_p, mapped, GD);
}